// wav_transform_80736795230663
// MI455X (gfx1250) — compile-verified
//
#include <hip/hip_runtime.h>

// Haar DWT level-1 on (32, 96, 224, 224) fp32 -> 4x (32, 96, 112, 112) fp32.
// Pure streaming kernel: non-temporal b128 loads/stores, division-free indexing.

typedef float v4f __attribute__((ext_vector_type(4)));

#define BC      (32 * 96)       // batch*channels = 3072
#define HI      224
#define WI      224
#define HO      112
#define WO      112
#define GROUPS  (WO / 4)        // 28 groups of 4 output columns per row
#define ROWS_PER_BLOCK 8
#define NOUT    (BC * HO * WO)  // elements per sub-band (38,535,168 < 2^31)

__global__ __launch_bounds__(GROUPS * ROWS_PER_BLOCK)   // 224 threads = 7 wave32
void haar_dwt_kernel(const float* __restrict__ x, float* __restrict__ out)
{
    const unsigned j4 = threadIdx.x;                              // 0..27
    const unsigned i  = blockIdx.x * ROWS_PER_BLOCK + threadIdx.y; // 0..111
    const unsigned bc = blockIdx.y;                               // 0..3071

    // 32-bit element offsets (max ~154M, fits easily)
    const unsigned in_base = bc * (HI * WI) + (2u * i) * WI + 8u * j4;

    const v4f* p0 = (const v4f*)(x + in_base);        // row 2i, cols [8j4, 8j4+8)
    const v4f* p1 = (const v4f*)(x + in_base + WI);   // row 2i+1

    v4f r0lo = __builtin_nontemporal_load(p0 + 0);
    v4f r0hi = __builtin_nontemporal_load(p0 + 1);
    v4f r1lo = __builtin_nontemporal_load(p1 + 0);
    v4f r1hi = __builtin_nontemporal_load(p1 + 1);

    // Output pixel k (k=0..3): a = row0 even col, b = row0 odd col,
    //                          c = row1 even col, d = row1 odd col.
    v4f a = { r0lo.x, r0lo.z, r0hi.x, r0hi.z };
    v4f b = { r0lo.y, r0lo.w, r0hi.y, r0hi.w };
    v4f c = { r1lo.x, r1lo.z, r1hi.x, r1hi.z };
    v4f d = { r1lo.y, r1lo.w, r1hi.y, r1hi.w };

    const float half = 0.5f;
    v4f apb = a + b, amb = a - b;
    v4f cpd = c + d, cmd = c - d;

    v4f LL = (apb + cpd) * half;
    v4f LH = (apb - cpd) * half;
    v4f HL = (amb + cmd) * half;
    v4f HH = (amb - cmd) * half;

    const unsigned ob = bc * (HO * WO) + i * WO + 4u * j4;

    __builtin_nontemporal_store(LL, (v4f*)(out + ob));
    __builtin_nontemporal_store(LH, (v4f*)(out + 1u * NOUT + ob));
    __builtin_nontemporal_store(HL, (v4f*)(out + 2u * NOUT + ob));
    __builtin_nontemporal_store(HH, (v4f*)(out + 3u * NOUT + ob));
}

extern "C" void kernel_launch(void* const* d_in, const int* in_sizes, int n_in,
                              void* d_out, int out_size, void* d_ws, size_t ws_size,
                              hipStream_t stream)
{
    (void)in_sizes; (void)n_in; (void)out_size; (void)d_ws; (void)ws_size;

    const float* x   = (const float*)d_in[0];
    float*       out = (float*)d_out;

    dim3 block(GROUPS, ROWS_PER_BLOCK);          // (28, 8) = 224 threads
    dim3 grid(HO / ROWS_PER_BLOCK, BC);          // (14, 3072)

    haar_dwt_kernel<<<grid, block, 0, stream>>>(x, out);
}